// OpenMaxLayer_14113262535035
// MI455X (gfx1250) — compile-verified
//
#include <hip/hip_runtime.h>
#include <math.h>

typedef float v2f __attribute__((ext_vector_type(2)));
typedef float v8f __attribute__((ext_vector_type(8)));

#define NB    16384   // B rows
#define NC    512     // classes
#define NF    1024    // feature dim
// ALPHA = 10 (w^10 computed by squaring)

// -------------------------------------------------------------------------
// Phase 1: Weibull factors. One wave (32 threads) handles 16 classes.
// d0sq for 16 classes = diagonal of diff * diff^T via V_WMMA_F32_16X16X4_F32.
// A 16x4 f32 layout: lane = M + 16*(K>=2), vgpr = K&1.
// B  4x16 f32 layout: lane = N + 16*(K>=2), vgpr = K&1  -> identical regs for
// a symmetric diff tile, so we pass `a` as both A and B.
// -------------------------------------------------------------------------
__global__ __launch_bounds__(32) void openmax_factor_kernel(
    const float* __restrict__ features,   // (B,F) - only row 0 used
    const float* __restrict__ mean_vecs,  // (C,F)
    const float* __restrict__ wparams,    // (C,3) : shape, loc, scale
    float* __restrict__ factor)           // (C,)  -> workspace
{
  const int lane = threadIdx.x & 31;
  const int row  = lane & 15;   // M (and N) index within the 16-class tile
  const int hi   = lane >> 4;   // 0 -> K{0,1}, 1 -> K{2,3}
  const int c0   = blockIdx.x * 16;

  const float* m = mean_vecs + (size_t)(c0 + row) * NF + 2 * hi;
  const float* x = features + 2 * hi;   // features[0]

  v8f acc = {};
  #pragma unroll 4
  for (int k = 0; k < NF; k += 4) {
    v2f a;
    a.x = m[k]     - x[k];      // feature k + 2*hi
    a.y = m[k + 1] - x[k + 1];  // feature k + 2*hi + 1
    // D = A * A^T + C  (8 args: neg_a, A, neg_b, B, c_mod, C, reuse_a, reuse_b)
    acc = __builtin_amdgcn_wmma_f32_16x16x4_f32(false, a, false, a,
                                                (short)0, acc, false, false);
  }

  // Extract diagonal element (M=N=j): vgpr = j&7, srcLane = j<8 ? j : j+16
  const int j = lane & 15;
  const int srcLane = (j < 8) ? j : (j + 16);
  float dsq = 0.0f;
  #pragma unroll
  for (int r = 0; r < 8; ++r) {
    float t = __shfl(acc[r], srcLane, 32);
    if ((j & 7) == r) dsq = t;
  }

  if (lane < 16) {
    const int c = c0 + j;
    const float cc    = wparams[3 * c + 0];
    const float loc   = wparams[3 * c + 1];
    const float scale = wparams[3 * c + 2];
    const float d0 = sqrtf(dsq);
    const float z  = fmaxf((d0 - loc) / scale, 0.0f);
    // w = 1 - exp(-z^cc);  z=0 -> z^cc=0 -> w=0 -> factor=1 (matches reference)
    const float w  = 1.0f - __expf(-__powf(z, cc));
    const float w2 = w * w;
    const float w4 = w2 * w2;
    const float w8 = w4 * w4;
    factor[c] = 1.0f - w8 * w2;   // 1 - w^10
  }
}

// -------------------------------------------------------------------------
// Phase 2: out = softmax(logits * factor, axis=1). Memory-bound (64 MB).
// One wave32 per row: 4 x float4 per lane covers 512 columns.
// Factors hoisted to registers (constant across rows). Shuffle reductions.
// -------------------------------------------------------------------------
__global__ __launch_bounds__(256) void openmax_softmax_kernel(
    const float* __restrict__ logits,   // (B,C)
    const float* __restrict__ factor,   // (C,)
    float* __restrict__ out,            // (B,C)
    int nrows)
{
  __shared__ float sfac[NC];
  for (int i = threadIdx.x; i < NC; i += 256) sfac[i] = factor[i];
  __syncthreads();

  const int lane = threadIdx.x & 31;
  const int wid  = threadIdx.x >> 5;

  // Per-lane factor registers: chunk q covers columns [q*128 + 4*lane, +4)
  float4 ff[4];
  #pragma unroll
  for (int q = 0; q < 4; ++q)
    ff[q] = ((const float4*)sfac)[q * 32 + lane];

  const int waves = gridDim.x * 8;
  for (int r = blockIdx.x * 8 + wid; r < nrows; r += waves) {
    const float4* lp = (const float4*)(logits + (size_t)r * NC);
    float4 v[4];
    #pragma unroll
    for (int q = 0; q < 4; ++q) {
      float4 t = lp[q * 32 + lane];               // global_load_b128
      v[q].x = t.x * ff[q].x;
      v[q].y = t.y * ff[q].y;
      v[q].z = t.z * ff[q].z;
      v[q].w = t.w * ff[q].w;
    }

    // Row max
    float mx = -3.402823466e38f;
    #pragma unroll
    for (int q = 0; q < 4; ++q)
      mx = fmaxf(mx, fmaxf(fmaxf(v[q].x, v[q].y), fmaxf(v[q].z, v[q].w)));
    #pragma unroll
    for (int off = 16; off >= 1; off >>= 1)
      mx = fmaxf(mx, __shfl_xor(mx, off, 32));

    // exp + row sum
    float s = 0.0f;
    #pragma unroll
    for (int q = 0; q < 4; ++q) {
      v[q].x = __expf(v[q].x - mx);
      v[q].y = __expf(v[q].y - mx);
      v[q].z = __expf(v[q].z - mx);
      v[q].w = __expf(v[q].w - mx);
      s += (v[q].x + v[q].y) + (v[q].z + v[q].w);
    }
    #pragma unroll
    for (int off = 16; off >= 1; off >>= 1)
      s += __shfl_xor(s, off, 32);

    const float inv = 1.0f / s;
    float4* op = (float4*)(out + (size_t)r * NC);
    #pragma unroll
    for (int q = 0; q < 4; ++q) {
      float4 o;
      o.x = v[q].x * inv;
      o.y = v[q].y * inv;
      o.z = v[q].z * inv;
      o.w = v[q].w * inv;
      op[q * 32 + lane] = o;                      // global_store_b128
    }
  }
}

extern "C" void kernel_launch(void* const* d_in, const int* in_sizes, int n_in,
                              void* d_out, int out_size, void* d_ws, size_t ws_size,
                              hipStream_t stream) {
  const float* logits    = (const float*)d_in[0];   // (B,C)
  const float* features  = (const float*)d_in[1];   // (B,F)
  const float* mean_vecs = (const float*)d_in[2];   // (C,F)
  const float* wparams   = (const float*)d_in[3];   // (C,3)
  float* out    = (float*)d_out;
  float* factor = (float*)d_ws;                     // 512 floats of scratch

  const int nrows = in_sizes[0] / NC;               // B

  // Phase 1: 512/16 = 32 class tiles, one wave each (WMMA f32 Gram diagonal)
  openmax_factor_kernel<<<NC / 16, 32, 0, stream>>>(features, mean_vecs,
                                                    wparams, factor);

  // Phase 2: 1024 blocks x 8 waves -> 8192 waves, ~2 rows per wave
  openmax_softmax_kernel<<<1024, 256, 0, stream>>>(logits, factor, out, nrows);
}